// GraphConv_65618510348365
// MI455X (gfx1250) — compile-verified
//
#include <hip/hip_runtime.h>

typedef __attribute__((ext_vector_type(2))) float v2f;
typedef __attribute__((ext_vector_type(8))) float v8f;

#define D_FEAT 64
#define LDS_STRIDE 65   // 64 + 1 pad: avoids 2-way bank conflict between lane halves

// ---------------------------------------------------------------------------
// z[m, :] = x[m, :] @ W   for m in [0, totalRows)   (totalRows = B*N, K=N=64)
// One wave32 computes a 16x64 output tile via V_WMMA_F32_16X16X4_F32.
// ---------------------------------------------------------------------------
__global__ void gcn_gemm_xw(const float* __restrict__ x,
                            const float* __restrict__ W,
                            float* __restrict__ z,
                            int totalRows) {
    __shared__ float Ws[D_FEAT * LDS_STRIDE];

    // Stage W (64x64 f32 = 16KB) into LDS with padded stride.
    for (int i = threadIdx.x; i < D_FEAT * D_FEAT; i += blockDim.x) {
        int r = i >> 6, c = i & 63;
        Ws[r * LDS_STRIDE + c] = W[i];
    }
    __syncthreads();

    const int wavesPerBlock = blockDim.x >> 5;
    const int tile = blockIdx.x * wavesPerBlock + (threadIdx.x >> 5);
    const int nTiles = totalRows >> 4;           // totalRows is a multiple of 16
    if (tile >= nTiles) return;

    const int lane = threadIdx.x & 31;
    const int half = lane >> 4;                  // 0: K0/K1  1: K2/K3 (A-frag ISA layout)
    const int l16  = lane & 15;
    const size_t m0 = (size_t)tile * 16;

    const float* __restrict__ arow = x + (m0 + (size_t)l16) * D_FEAT;

    v8f acc0 = {}, acc1 = {}, acc2 = {}, acc3 = {};

    #pragma unroll
    for (int k0 = 0; k0 < D_FEAT; k0 += 4) {
        // A fragment: 16x4 f32. lane<16 holds (K0,K1), lane>=16 holds (K2,K3).
        v2f a;
        a.x = arow[k0 + 2 * half + 0];
        a.y = arow[k0 + 2 * half + 1];

        // B fragments: 4x16 f32 per N-tile, mirrored layout (rows K0/K1 | K2/K3).
        #pragma unroll
        for (int j = 0; j < 4; ++j) {
            v2f b;
            const int col = j * 16 + l16;
            b.x = Ws[(k0 + 2 * half + 0) * LDS_STRIDE + col];
            b.y = Ws[(k0 + 2 * half + 1) * LDS_STRIDE + col];
            v8f* accp = (j == 0) ? &acc0 : (j == 1) ? &acc1 : (j == 2) ? &acc2 : &acc3;
            *accp = __builtin_amdgcn_wmma_f32_16x16x4_f32(
                false, a, false, b, (short)0, *accp, false, false);
        }
    }

    // D layout: VGPR r -> row m0+r (lanes 0-15) / m0+8+r (lanes 16-31), col = l16.
    #pragma unroll
    for (int j = 0; j < 4; ++j) {
        const v8f acc = (j == 0) ? acc0 : (j == 1) ? acc1 : (j == 2) ? acc2 : acc3;
        #pragma unroll
        for (int r = 0; r < 8; ++r) {
            const size_t row = m0 + (size_t)r + 8 * half;
            z[row * D_FEAT + j * 16 + l16] = acc[r];
        }
    }
}

// ---------------------------------------------------------------------------
// Zero-init d_out (float4 granularity).
// ---------------------------------------------------------------------------
__global__ void zero_out4(float4* __restrict__ p, int n4) {
    int i = blockIdx.x * blockDim.x + threadIdx.x;
    if (i < n4) p[i] = make_float4(0.f, 0.f, 0.f, 0.f);
}

// ---------------------------------------------------------------------------
// out[b][rows[e]][:] += a_vals[e] * z[b][cols[e]][:]
// One wave32 per (edge, batch); 2 features per lane.
// ---------------------------------------------------------------------------
__global__ void gcn_scatter(const float* __restrict__ z,
                            const int* __restrict__ rows,
                            const int* __restrict__ cols,
                            const float* __restrict__ avals,
                            float* __restrict__ out,
                            int E, int N) {
    const int wid  = (blockIdx.x * blockDim.x + threadIdx.x) >> 5;
    const int lane = threadIdx.x & 31;
    if (wid >= E * 4) return;

    const int b = wid / E;        // batch
    const int e = wid - b * E;    // edge

    const int   r = rows[e];      // wave-uniform loads (broadcast from cache)
    const int   c = cols[e];
    const float a = avals[e];

    const float* __restrict__ zp = z   + ((size_t)b * N + (size_t)c) * D_FEAT;
    float* __restrict__       op = out + ((size_t)b * N + (size_t)r) * D_FEAT;

    const float v0 = zp[lane]      * a;
    const float v1 = zp[lane + 32] * a;
    atomicAdd(op + lane,      v0);  // -> global_atomic_add_f32
    atomicAdd(op + lane + 32, v1);
}

// ---------------------------------------------------------------------------
extern "C" void kernel_launch(void* const* d_in, const int* in_sizes, int n_in,
                              void* d_out, int out_size, void* d_ws, size_t ws_size,
                              hipStream_t stream) {
    const float* x     = (const float*)d_in[0];   // [B, N, 64] f32
    const float* W     = (const float*)d_in[1];   // [64, 64]   f32
    const int*   rows  = (const int*)  d_in[2];   // [E] i32
    const int*   cols  = (const int*)  d_in[3];   // [E] i32
    const float* avals = (const float*)d_in[4];   // [E] f32

    float* out = (float*)d_out;
    float* z   = (float*)d_ws;                    // [B, N, 64] f32 workspace (51.2 MB)

    const int E         = in_sizes[2];
    const int totalRows = in_sizes[0] / D_FEAT;   // B * N
    const int N         = totalRows / 4;          // B == 4

    // 1) z = x @ W  (associativity: (A x) W == A (x W))
    {
        const int tiles  = totalRows >> 4;        // 16-row tiles
        const int wpb    = 8;                     // 256 threads = 8 wave32
        const int blocks = (tiles + wpb - 1) / wpb;
        gcn_gemm_xw<<<blocks, 256, 0, stream>>>(x, W, z, totalRows);
    }

    // 2) out = 0
    {
        const int n4 = out_size / 4;
        zero_out4<<<(n4 + 255) / 256, 256, 0, stream>>>((float4*)d_out, n4);
    }

    // 3) out[b][rows[e]] += a_vals[e] * z[b][cols[e]]
    {
        const long waves   = (long)E * 4;
        const long threads = waves * 32;
        const int  blocks  = (int)((threads + 255) / 256);
        gcn_scatter<<<blocks, 256, 0, stream>>>(z, rows, cols, avals, out, E, N);
    }
}